// SparseConv3d_40570261078329
// MI455X (gfx1250) — compile-verified
//
#include <hip/hip_runtime.h>

typedef __attribute__((ext_vector_type(2))) float v2f;
typedef __attribute__((ext_vector_type(8))) float v8f;

#define CIN 32
#define COUT 32
#define EPSV 1e-5f
#define TILE 16                    // points per wave (WMMA M dim)
#define NWAVES 8                   // waves per workgroup (256 threads, wave32)
#define PTS_PER_WG (TILE * NWAVES) // 128 points per workgroup
#define LDS_STRIDE 36              // dwords per staged row: 16B-aligned, bank-conflict-free
#define TILE_DW (TILE * LDS_STRIDE)

// Async per-lane 16B copy global -> LDS (CDNA5 GLOBAL_LOAD_ASYNC_TO_LDS_B128, ASYNCcnt).
__device__ __forceinline__ void async_gather_b128(unsigned lds_byte_off, unsigned long long ga) {
    asm volatile("global_load_async_to_lds_b128 %0, %1, off"
                 :: "v"(lds_byte_off), "v"(ga)
                 : "memory");
}
template <int K>
__device__ __forceinline__ void wait_async() {
    asm volatile("s_wait_asynccnt %0" :: "i"(K) : "memory");
}

// Issue the 16-row (128B each) gather for one kernel offset, fully branchless:
// missing neighbors (ri == n) are redirected to the zero pad row via mask select.
__device__ __forceinline__ void issue_gather(int rv, unsigned tb,
                                             const float* __restrict__ fin,
                                             const float* __restrict__ zrow,
                                             int n, int lane) {
    #pragma unroll
    for (int it = 0; it < 4; ++it) {
        int q  = it * 32 + lane;  // 0..127
        int r  = q >> 3;          // row 0..15
        int f4 = q & 7;           // float4 within row
        int ri = __shfl(rv, r, 32);
        unsigned long long av = (unsigned long long)(uintptr_t)(fin + (size_t)ri * CIN + f4 * 4);
        unsigned long long az = (unsigned long long)(uintptr_t)(zrow + f4 * 4);
        unsigned long long mk = (ri < n) ? ~0ull : 0ull;
        unsigned long long ga = (av & mk) | (az & ~mk);
        async_gather_b128(tb + (unsigned)((r * LDS_STRIDE + f4 * 4) * 4), ga);
    }
}

// 16 x V_WMMA_F32_16X16X4_F32 for one offset j: A from LDS tile, B from LDS weight pairs.
__device__ __forceinline__ void compute_j(int j, const float* __restrict__ T,
                                          const v2f* __restrict__ wPairs,
                                          int m, int hg, v8f& acc_lo, v8f& acc_hi) {
    #pragma unroll
    for (int k = 0; k < 8; ++k) {
        // A fragment (16x4 f32): lane(m,hg) holds f[row m, K=4k+2hg .. +1]
        v2f a = *(const v2f*)(T + m * LDS_STRIDE + k * 4 + 2 * hg);
        // B fragment (4x16 f32): pre-paired rows, XOR-16 swizzled by hg (bank split 0-31/32-63)
        int base = ((j * 8 + k) * 2 + hg) * 32;
        v2f blo = wPairs[base + m + 16 * hg];        // col m
        v2f bhi = wPairs[base + m + 16 * (1 - hg)];  // col m+16
        acc_lo = __builtin_amdgcn_wmma_f32_16x16x4_f32(
            false, a, false, blo, (short)0, acc_lo, false, false);
        acc_hi = __builtin_amdgcn_wmma_f32_16x16x4_f32(
            false, a, false, bhi, (short)0, acc_hi, false, false);
    }
}

// One factorized sparse-conv stage: fout[p,:] = sum_j W[j]^T @ fin[idx[j,p],:]
// Per wave: async DMA gathers double-buffered against 48 f32 WMMAs.
__global__ __launch_bounds__(256) void sconv_wmma_kernel(
    const float* __restrict__ fin,      // [n,32]
    const float* __restrict__ W,        // [3,32,32]
    const int*   __restrict__ idxA,     // [3,n], value n => missing neighbor (zero row)
    const float* __restrict__ zrow,     // [32] zeroed pad row (the reference's fpad[n])
    float*       __restrict__ fout,     // [n,32]
    float*       __restrict__ partials, // [gridDim.x,64] (sum[32], sumsq[32]) when do_stats
    int n, int do_stats)
{
    __shared__ float tileMem[NWAVES][2][TILE_DW];  // 36.9 KB: double-buffered gather tiles
    __shared__ v2f   wPairs[48 * 32];              // 12.3 KB: all of W as row pairs

    const int lane = threadIdx.x & 31;
    const int wave = threadIdx.x >> 5;
    const int m    = lane & 15;   // row-in-tile (M) / column (N) index for fragments
    const int hg   = lane >> 4;   // half-group: selects K pair / M-half per ISA layout
    const int tile_base = (blockIdx.x * NWAVES + wave) * TILE;
    float* B0 = tileMem[wave][0];
    float* B1 = tileMem[wave][1];
    const unsigned Tb0 = (unsigned)(uintptr_t)B0;  // flat addr low 32 bits = LDS offset
    const unsigned Tb1 = (unsigned)(uintptr_t)B1;

    // Branchless neighbor indices for point (tile_base + m); lanes 16..31 mirror 0..15.
    const int p  = tile_base + m;
    const int pc = (p < n) ? p : (n - 1);
    int rIdx[3];
    #pragma unroll
    for (int j = 0; j < 3; ++j) {
        int v = idxA[(size_t)j * (size_t)n + pc];
        rIdx[j] = (p < n) ? v : n;
    }

    // Kick off the first two offsets' gathers before touching weights.
    issue_gather(rIdx[0], Tb0, fin, zrow, n, lane);
    issue_gather(rIdx[1], Tb1, fin, zrow, n, lane);

    // Cooperative weight staging (runs under the async gathers): 1536 float2 entries.
    // Entry (j,k,hg,slot): col = slot ^ (hg<<4); holds (W[4k+2hg][col], W[4k+2hg+1][col]).
    #pragma unroll
    for (int t = 0; t < 6; ++t) {
        int e    = threadIdx.x + t * 256;  // 0..1535
        int slot = e & 31;
        int rem  = e >> 5;                 // 0..47 = (j*8+k)*2+hg
        int hgw  = rem & 1;
        int kj   = rem >> 1;
        int kk   = kj & 7;
        int jj   = kj >> 3;
        int col  = slot ^ (hgw << 4);
        int row0 = kk * 4 + hgw * 2;
        const float* Wj = W + jj * CIN * COUT;
        v2f pr;
        pr.x = Wj[(row0 + 0) * COUT + col];
        pr.y = Wj[(row0 + 1) * COUT + col];
        wPairs[rem * 32 + slot] = pr;
    }
    __syncthreads();

    v8f acc_lo = {}; // D rows: M = r + 8*hg, col = m      (channels 0..15)
    v8f acc_hi = {}; //                       col = m + 16 (channels 16..31)

    // Software pipeline on in-order ASYNCcnt: compute j while the next gather lands.
    wait_async<4>();                                   // j0 tile resident
    compute_j(0, B0, wPairs, m, hg, acc_lo, acc_hi);
    issue_gather(rIdx[2], Tb0, fin, zrow, n, lane);    // j0 frags consumed; reuse buf0
    wait_async<4>();                                   // j1 tile resident
    compute_j(1, B1, wPairs, m, hg, acc_lo, acc_hi);
    wait_async<0>();                                   // j2 tile resident
    compute_j(2, B0, wPairs, m, hg, acc_lo, acc_hi);
    __builtin_amdgcn_wave_barrier();

    // Transpose accumulators through LDS (buf1 is free) -> contiguous 128B row stores.
    #pragma unroll
    for (int r = 0; r < 8; ++r) {
        int row = r + 8 * hg;
        B1[row * LDS_STRIDE + m]      = acc_lo[r];
        B1[row * LDS_STRIDE + 16 + m] = acc_hi[r];
    }
    __builtin_amdgcn_wave_barrier();

    #pragma unroll
    for (int it = 0; it < 4; ++it) {
        int q  = it * 32 + lane;
        int r  = q >> 3;
        int f4 = q & 7;
        int pp = tile_base + r;
        if (pp < n)
            *(float4*)(fout + (size_t)pp * COUT + f4 * 4) =
                *(const float4*)(B1 + r * LDS_STRIDE + f4 * 4);
    }

    if (do_stats) {
        // per-lane partial sum / sumsq over its 8 accumulator rows (zeros past n are harmless)
        float s0 = 0.f, q0 = 0.f, s1 = 0.f, q1 = 0.f;
        #pragma unroll
        for (int r = 0; r < 8; ++r) {
            float a = acc_lo[r]; s0 += a; q0 += a * a;
            float b = acc_hi[r]; s1 += b; q1 += b * b;
        }
        // stat scratch reuses buf0 region (per-wave, all prior reads consumed)
        float4* myStat = (float4*)(&tileMem[wave][0][0]);
        myStat[lane] = make_float4(s0, q0, s1, q1);
        __syncthreads();
        if (threadIdx.x < 32) {           // deterministic fixed-order block reduction
            int c = threadIdx.x;
            float sum = 0.f, ssq = 0.f;
            if (c < 16) {
                for (int w = 0; w < NWAVES; ++w) {
                    const float4* st = (const float4*)(&tileMem[w][0][0]);
                    float4 u0 = st[c];
                    float4 u1 = st[c + 16];
                    sum += u0.x + u1.x;
                    ssq += u0.y + u1.y;
                }
            } else {
                int mm = c - 16;
                for (int w = 0; w < NWAVES; ++w) {
                    const float4* st = (const float4*)(&tileMem[w][0][0]);
                    float4 u0 = st[mm];
                    float4 u1 = st[mm + 16];
                    sum += u0.z + u1.z;
                    ssq += u0.w + u1.w;
                }
            }
            partials[(size_t)blockIdx.x * 64 + c]      = sum;
            partials[(size_t)blockIdx.x * 64 + 32 + c] = ssq;
        }
    }
}

// Zero the pad row every launch (workspace is not re-zeroed by the harness).
__global__ void init_zrow_kernel(float* __restrict__ zrow) {
    zrow[threadIdx.x] = 0.f;
}

// Fold BN stats + gamma/beta into per-channel scale & bias (single wave, fixed order).
__global__ void bn_finalize_kernel(const float* __restrict__ partials, int nblocks,
                                   const float* __restrict__ gamma,
                                   const float* __restrict__ beta,
                                   float* __restrict__ sb, int n)
{
    int c = threadIdx.x;
    if (c < 32) {
        float sum = 0.f, ssq = 0.f;
        for (int w = 0; w < nblocks; ++w) {
            sum += partials[(size_t)w * 64 + c];
            ssq += partials[(size_t)w * 64 + 32 + c];
        }
        float inv_n = 1.0f / (float)n;
        float mean  = sum * inv_n;
        float var   = ssq * inv_n - mean * mean;
        float sc    = gamma[c] * rsqrtf(var + EPSV);
        sb[c]       = sc;
        sb[32 + c]  = beta[c] - mean * sc;
    }
}

// In-place BN + ReLU over [n,32], vectorized float4.
__global__ __launch_bounds__(256) void bn_relu_kernel(
    float* __restrict__ h, const float* __restrict__ sb, int total4)
{
    __shared__ float s_sc[32], s_bi[32];
    if (threadIdx.x < 32) {
        s_sc[threadIdx.x] = sb[threadIdx.x];
        s_bi[threadIdx.x] = sb[32 + threadIdx.x];
    }
    __syncthreads();
    int stride = gridDim.x * blockDim.x;
    for (int i = blockIdx.x * blockDim.x + threadIdx.x; i < total4; i += stride) {
        float4 v = ((float4*)h)[i];
        int c0 = (i * 4) & 31;
        v.x = fmaxf(v.x * s_sc[c0 + 0] + s_bi[c0 + 0], 0.f);
        v.y = fmaxf(v.y * s_sc[c0 + 1] + s_bi[c0 + 1], 0.f);
        v.z = fmaxf(v.z * s_sc[c0 + 2] + s_bi[c0 + 2], 0.f);
        v.w = fmaxf(v.w * s_sc[c0 + 3] + s_bi[c0 + 3], 0.f);
        ((float4*)h)[i] = v;
    }
}

extern "C" void kernel_launch(void* const* d_in, const int* in_sizes, int n_in,
                              void* d_out, int out_size, void* d_ws, size_t ws_size,
                              hipStream_t stream) {
    (void)n_in; (void)out_size; (void)ws_size;
    const float* feats = (const float*)d_in[0];
    const float* W1    = (const float*)d_in[1];
    const float* W2    = (const float*)d_in[2];
    const float* W3    = (const float*)d_in[3];
    const float* gamma = (const float*)d_in[4];
    const float* beta  = (const float*)d_in[5];
    const int*   nbr   = (const int*)d_in[6];   // [3 axes][3 offsets][n]
    float* out = (float*)d_out;

    const int n   = in_sizes[0] / CIN;
    const int nWG = (n + PTS_PER_WG - 1) / PTS_PER_WG;

    float* hA       = (float*)d_ws;                       // [n,32]
    float* hB       = hA + (size_t)n * COUT;              // [n,32]
    float* partials = hB + (size_t)n * COUT;              // [nWG,64]
    float* sb       = partials + (size_t)nWG * 64;        // [64] scale+bias
    float* zrow     = sb + 64;                            // [32] zero pad row

    const size_t axStride = (size_t)3 * (size_t)n;        // stride of nbr_idx axis dim

    init_zrow_kernel<<<1, 32, 0, stream>>>(zrow);
    // conv1: kernel (1,1,3) -> axis 2; conv2: (1,3,1) -> axis 1; conv3: (3,1,1) -> axis 0
    sconv_wmma_kernel<<<nWG, 256, 0, stream>>>(feats, W1, nbr + 2 * axStride, zrow, hA,
                                               nullptr, n, 0);
    sconv_wmma_kernel<<<nWG, 256, 0, stream>>>(hA, W2, nbr + 1 * axStride, zrow, hB,
                                               nullptr, n, 0);
    sconv_wmma_kernel<<<nWG, 256, 0, stream>>>(hB, W3, nbr /* axis 0 */, zrow, out,
                                               partials, n, 1);
    bn_finalize_kernel<<<1, 32, 0, stream>>>(partials, nWG, gamma, beta, sb, n);
    bn_relu_kernel<<<1024, 256, 0, stream>>>(out, sb, n * COUT / 4);
}